// CgpHmmCell_6545530159760
// MI455X (gfx1250) — compile-verified
//
#include <hip/hip_runtime.h>

// ---------------------------------------------------------------------------
// HMM forward scan, fused, using V_WMMA_F32_16X16X4_F32 on gfx1250 (wave32).
//   alpha_t = E_t ⊙ (alpha_{t-1} @ A),  E_t = x_t @ Bm^T
// Computed in transposed wave-matrix form:
//   E^T     = Bm(5x4)  x  x_t^T(4x16)      -> one 16x16x4 f32 WMMA (K=4=ALPHABET)
//   alphaN^T= A^T(5x5) x  alpha^T(5x16)    -> two accumulating WMMAs (K=4 + K=1)
// One wave handles 2 independent chains of 16 batch columns each.
// ---------------------------------------------------------------------------

typedef float v2f __attribute__((ext_vector_type(2)));
typedef float v8f __attribute__((ext_vector_type(8)));

#define HMM_S 5
#define ALPH  4
#define BATCH 2048
#define TLEN  4096
#define NC    2      // independent chains per wave (ILP on the serial scan)

static __device__ __forceinline__ v8f wmma4(v2f a, v2f b, v8f c) {
#if defined(__HIP_DEVICE_COMPILE__)
#if !__has_builtin(__builtin_amdgcn_wmma_f32_16x16x4_f32)
#error "gfx1250 device pass: __builtin_amdgcn_wmma_f32_16x16x4_f32 not available"
#endif
  // (neg_a, A, neg_b, B, c_mod, C, reuse_a, reuse_b)
  return __builtin_amdgcn_wmma_f32_16x16x4_f32(false, a, false, b, (short)0, c,
                                               false, false);
#else
  (void)a; (void)b;
  return c;  // host pass stub (never executed)
#endif
}

__global__ __launch_bounds__(32) void hmm_forward_kernel(
    const float* __restrict__ x,    // (B, T, 4)
    const float* __restrict__ tk,   // (5, 5) transition logits
    const float* __restrict__ ek,   // (5, 4) emission logits
    float* __restrict__ out)        // (B, T, 5)
{
  const int lane = threadIdx.x & 31;
  const int col  = lane & 15;   // N / M index within tile
  const int half = lane >> 4;   // K half (0: K=0,1 ; 1: K=2,3)

  // ---- softmax(transition) -> A[5][5], softmax(emission) -> Bm[5][4] ------
  float A[HMM_S][HMM_S];
  #pragma unroll
  for (int i = 0; i < HMM_S; ++i) {
    float r[HMM_S]; float m = -3.0e38f;
    #pragma unroll
    for (int j = 0; j < HMM_S; ++j) { r[j] = tk[i * HMM_S + j]; m = fmaxf(m, r[j]); }
    float s = 0.f;
    #pragma unroll
    for (int j = 0; j < HMM_S; ++j) { r[j] = __expf(r[j] - m); s += r[j]; }
    const float inv = 1.f / s;
    #pragma unroll
    for (int j = 0; j < HMM_S; ++j) A[i][j] = r[j] * inv;
  }
  float Bm[HMM_S][ALPH];
  #pragma unroll
  for (int i = 0; i < HMM_S; ++i) {
    float r[ALPH]; float m = -3.0e38f;
    #pragma unroll
    for (int j = 0; j < ALPH; ++j) { r[j] = ek[i * ALPH + j]; m = fmaxf(m, r[j]); }
    float s = 0.f;
    #pragma unroll
    for (int j = 0; j < ALPH; ++j) { r[j] = __expf(r[j] - m); s += r[j]; }
    const float inv = 1.f / s;
    #pragma unroll
    for (int j = 0; j < ALPH; ++j) Bm[i][j] = r[j] * inv;
  }

  // ---- constant A-operands (16x4 f32 layout: M=col, K split across halves) -
  const bool rowOK = (col < HMM_S);
  v2f bmA;                       // emission:   Amat[m][k] = Bm[m][k]
  bmA.x  = rowOK ? Bm[col][2 * half + 0] : 0.f;
  bmA.y  = rowOK ? Bm[col][2 * half + 1] : 0.f;
  v2f at03;                      // transition: Amat[m][k] = A[k][m], k=0..3
  at03.x = rowOK ? A[2 * half + 0][col] : 0.f;
  at03.y = rowOK ? A[2 * half + 1][col] : 0.f;
  v2f at4;                       // transition fix-up: K slot 0 carries k=4
  at4.x  = (rowOK && half == 0) ? A[4][col] : 0.f;
  at4.y  = 0.f;

  const v8f zero8 = {0.f, 0.f, 0.f, 0.f, 0.f, 0.f, 0.f, 0.f};

  // ---- per-chain bases -----------------------------------------------------
  const int tile0 = blockIdx.x * NC;
  const float* xp0[NC];
  float*       op0[NC];
  #pragma unroll
  for (int c = 0; c < NC; ++c) {
    const size_t b = (size_t)(tile0 + c) * 16 + (size_t)col;  // batch row
    xp0[c] = x   + b * (size_t)TLEN * ALPH + 2 * half;        // float2 per lane
    op0[c] = out + b * (size_t)TLEN * HMM_S;                  // lanes 0-15 store
  }

  float a0[NC], a1[NC], a2[NC], a3[NC], a4[NC];
  v2f xnext[NC];
  #pragma unroll
  for (int c = 0; c < NC; ++c) xnext[c] = *(const v2f*)(xp0[c]);   // t=0

  // ---- t = 0: alpha = [E0[:,0], 0,0,0,0] ----------------------------------
  #pragma unroll
  for (int c = 0; c < NC; ++c) {
    v8f e = wmma4(bmA, xnext[c], zero8);
    a0[c] = e[0];
    a1[c] = 0.f; a2[c] = 0.f; a3[c] = 0.f; a4[c] = 0.f;
    xnext[c] = *(const v2f*)(xp0[c] + (size_t)1 * ALPH);           // prefetch t=1
  }
  if (lane < 16) {
    #pragma unroll
    for (int c = 0; c < NC; ++c) {
      float* p = op0[c];
      p[0] = a0[c]; p[1] = a1[c]; p[2] = a2[c]; p[3] = a3[c]; p[4] = a4[c];
    }
  }

  // ---- main scan -----------------------------------------------------------
  for (int t = 1; t < TLEN; ++t) {
    v2f xc[NC];
    #pragma unroll
    for (int c = 0; c < NC; ++c) xc[c] = xnext[c];
    if (t + 1 < TLEN) {
      #pragma unroll
      for (int c = 0; c < NC; ++c)
        xnext[c] = *(const v2f*)(xp0[c] + (size_t)(t + 1) * ALPH);
    }

    #pragma unroll
    for (int c = 0; c < NC; ++c) {
      // emission tile: e[r] = E_t[state r][batch col]   (lanes 0-15)
      v8f e = wmma4(bmA, xc[c], zero8);

      // alpha^T B-operand (4x16): lower lanes carry k=0,1; upper carry k=2,3
      const float s2 = __shfl(a2[c], col, 32);
      const float s3 = __shfl(a3[c], col, 32);
      v2f b03, b4;
      b03.x = half ? s2  : a0[c];
      b03.y = half ? s3  : a1[c];
      b4.x  = half ? 0.f : a4[c];    // k=4 row in K slot 0
      b4.y  = 0.f;

      v8f d = wmma4(at4, b4, zero8);     // += A[4][m] * alpha[.][4]
      d     = wmma4(at03, b03, d);       // += sum_{k<4} A[k][m] * alpha[.][k]

      a0[c] = e[0] * d[0];
      a1[c] = e[1] * d[1];
      a2[c] = e[2] * d[2];
      a3[c] = e[3] * d[3];
      a4[c] = e[4] * d[4];
    }

    if (lane < 16) {
      #pragma unroll
      for (int c = 0; c < NC; ++c) {
        float* p = op0[c] + (size_t)t * HMM_S;
        p[0] = a0[c]; p[1] = a1[c]; p[2] = a2[c]; p[3] = a3[c]; p[4] = a4[c];
      }
    }
  }
}

extern "C" void kernel_launch(void* const* d_in, const int* in_sizes, int n_in,
                              void* d_out, int out_size, void* d_ws, size_t ws_size,
                              hipStream_t stream) {
  (void)in_sizes; (void)n_in; (void)d_ws; (void)ws_size; (void)out_size;
  const float* x  = (const float*)d_in[0];   // (2048, 4096, 4) f32
  const float* tk = (const float*)d_in[1];   // (5, 5) f32
  const float* ek = (const float*)d_in[2];   // (5, 4) f32
  float* out = (float*)d_out;                // (2048, 4096, 5) f32

  dim3 grid(BATCH / (16 * NC));   // 64 single-wave workgroups, 2 chains each
  dim3 block(32);
  hipLaunchKernelGGL(hmm_forward_kernel, grid, block, 0, stream, x, tk, ek, out);
}